// DiffusionLoss_53300544143992
// MI455X (gfx1250) — compile-verified
//
#include <hip/hip_runtime.h>
#include <hip/hip_bf16.h>
#include <math.h>

typedef __attribute__((ext_vector_type(2))) float v2f;
typedef __attribute__((ext_vector_type(8))) float v8f;

#define SIGMA_DATA 16.0f
#define ALPHA_BOND 1.0f
#define B_DIM 4
#define N_DIM 2048
#define NT (N_DIM / 16)            // 128 tiles per row
#define TILES_PER_B (NT * NT)      // 16384

// workspace float layout
#define WS_LOSS1   0
#define WS_BOND    1
#define WS_BATCH0  16
#define WS_BSTRIDE 32
#define O_SW   0
#define O_SP   1
#define O_SG   4
#define O_SPG  7
#define O_R    16
#define O_MUG  25
#define O_MUP  28
#define WS_FLOATS (WS_BATCH0 + B_DIM * WS_BSTRIDE)

__global__ __launch_bounds__(256) void init_ws(float* ws) {
    int t = threadIdx.x;
    if (t < WS_FLOATS) ws[t] = 0.0f;
}

// One-pass weighted moments: Sw, Sum(w*xp), Sum(w*xg), Sum(w * xp_i * xg_j)
__global__ __launch_bounds__(256) void stats_kernel(const float* __restrict__ xp,
                                                    const float* __restrict__ xg,
                                                    const float* __restrict__ w,
                                                    float* ws) {
    int b = blockIdx.x >> 3;                       // 8 blocks per batch (2048/256)
    int n = ((blockIdx.x & 7) << 8) + threadIdx.x;
    const float* p = xp + ((size_t)b * N_DIM + n) * 3;
    const float* g = xg + ((size_t)b * N_DIM + n) * 3;
    float wv = w[(size_t)b * N_DIM + n];
    float px = p[0], py = p[1], pz = p[2];
    float gx = g[0], gy = g[1], gz = g[2];

    float vals[16];
    vals[0] = wv;
    vals[1] = wv * px; vals[2] = wv * py; vals[3] = wv * pz;
    vals[4] = wv * gx; vals[5] = wv * gy; vals[6] = wv * gz;
    vals[7]  = wv * px * gx; vals[8]  = wv * px * gy; vals[9]  = wv * px * gz;
    vals[10] = wv * py * gx; vals[11] = wv * py * gy; vals[12] = wv * py * gz;
    vals[13] = wv * pz * gx; vals[14] = wv * pz * gy; vals[15] = wv * pz * gz;

    __shared__ float sd[256];
    float* base = ws + WS_BATCH0 + b * WS_BSTRIDE;
    for (int k = 0; k < 16; ++k) {
        sd[threadIdx.x] = vals[k];
        __syncthreads();
        for (int off = 128; off > 0; off >>= 1) {
            if (threadIdx.x < off) sd[threadIdx.x] += sd[threadIdx.x + off];
            __syncthreads();
        }
        if (threadIdx.x == 0) atomicAdd(base + k, sd[0]);
        __syncthreads();
    }
}

__device__ inline void jacobi3(float A[3][3], float V[3][3]) {
    for (int i = 0; i < 3; i++)
        for (int j = 0; j < 3; j++) V[i][j] = (i == j) ? 1.0f : 0.0f;
    const int pqs[3][2] = {{0,1},{0,2},{1,2}};
    for (int sweep = 0; sweep < 14; ++sweep) {
        for (int k = 0; k < 3; ++k) {
            int p = pqs[k][0], q = pqs[k][1];
            float apq = A[p][q];
            if (fabsf(apq) < 1e-24f) continue;
            float theta = (A[q][q] - A[p][p]) / (2.0f * apq);
            float t = ((theta >= 0.0f) ? 1.0f : -1.0f) /
                      (fabsf(theta) + sqrtf(theta * theta + 1.0f));
            float c = 1.0f / sqrtf(t * t + 1.0f);
            float s = t * c;
            for (int i = 0; i < 3; i++) {
                float aip = A[i][p], aiq = A[i][q];
                A[i][p] = c * aip - s * aiq;
                A[i][q] = s * aip + c * aiq;
            }
            for (int i = 0; i < 3; i++) {
                float api = A[p][i], aqi = A[q][i];
                A[p][i] = c * api - s * aqi;
                A[q][i] = s * api + c * aqi;
            }
            for (int i = 0; i < 3; i++) {
                float vip = V[i][p], viq = V[i][q];
                V[i][p] = c * vip - s * viq;
                V[i][q] = s * vip + c * viq;
            }
        }
    }
}

// Per-batch 3x3 Kabsch: M = Spg - Sp*Sg^T/Sw ; SVD; R = U*diag(1,1,det fix)*V^T
__global__ void svd_kernel(float* ws) {
    int b = threadIdx.x;
    if (b >= B_DIM) return;
    float* base = ws + WS_BATCH0 + b * WS_BSTRIDE;
    float Sw = base[O_SW];
    float Sp[3], Sg[3];
    for (int i = 0; i < 3; i++) { Sp[i] = base[O_SP + i]; Sg[i] = base[O_SG + i]; }
    float M[3][3];
    for (int i = 0; i < 3; i++)
        for (int j = 0; j < 3; j++)
            M[i][j] = base[O_SPG + 3 * i + j] - Sp[i] * Sg[j] / Sw;

    float A[3][3];
    for (int i = 0; i < 3; i++)
        for (int j = 0; j < 3; j++) {
            float s = 0.0f;
            for (int k = 0; k < 3; k++) s += M[k][i] * M[k][j];
            A[i][j] = s;
        }
    float V[3][3];
    jacobi3(A, V);
    float d[3] = {A[0][0], A[1][1], A[2][2]};
    // sort eigenvalues descending, permute V columns
    for (int a = 0; a < 2; a++) {
        int mx = a;
        for (int c = a + 1; c < 3; c++) if (d[c] > d[mx]) mx = c;
        if (mx != a) {
            float tmp = d[a]; d[a] = d[mx]; d[mx] = tmp;
            for (int i = 0; i < 3; i++) {
                float tv = V[i][a]; V[i][a] = V[i][mx]; V[i][mx] = tv;
            }
        }
    }
    float U[3][3];
    for (int k = 0; k < 2; k++) {
        float u[3];
        for (int i = 0; i < 3; i++)
            u[i] = M[i][0] * V[0][k] + M[i][1] * V[1][k] + M[i][2] * V[2][k];
        if (k == 1) {
            float dp = u[0]*U[0][0] + u[1]*U[1][0] + u[2]*U[2][0];
            for (int i = 0; i < 3; i++) u[i] -= dp * U[i][0];
        }
        float nn = sqrtf(u[0]*u[0] + u[1]*u[1] + u[2]*u[2]);
        if (nn > 1e-12f) {
            for (int i = 0; i < 3; i++) U[i][k] = u[i] / nn;
        } else {
            for (int i = 0; i < 3; i++) U[i][k] = (i == k) ? 1.0f : 0.0f;
        }
    }
    // u2 = u0 x u1  (det(U)=+1)
    U[0][2] = U[1][0]*U[2][1] - U[2][0]*U[1][1];
    U[1][2] = U[2][0]*U[0][1] - U[0][0]*U[2][1];
    U[2][2] = U[0][0]*U[1][1] - U[1][0]*U[0][1];

    float R[3][3];
    for (int i = 0; i < 3; i++)
        for (int j = 0; j < 3; j++)
            R[i][j] = U[i][0]*V[j][0] + U[i][1]*V[j][1] + U[i][2]*V[j][2];
    float det = R[0][0]*(R[1][1]*R[2][2]-R[1][2]*R[2][1])
              - R[0][1]*(R[1][0]*R[2][2]-R[1][2]*R[2][0])
              + R[0][2]*(R[1][0]*R[2][1]-R[1][1]*R[2][0]);
    if (det < 0.0f) {
        for (int i = 0; i < 3; i++)
            for (int j = 0; j < 3; j++)
                R[i][j] = U[i][0]*V[j][0] + U[i][1]*V[j][1] - U[i][2]*V[j][2];
    }
    for (int i = 0; i < 3; i++)
        for (int j = 0; j < 3; j++) base[O_R + 3*i + j] = R[i][j];
    for (int i = 0; i < 3; i++) {
        base[O_MUG + i] = Sg[i] / Sw;
        base[O_MUP + i] = Sp[i] / Sw;
    }
}

// First loss term: sum_n w * || xpred - (R*(xgt - mu_g) + mu_p) ||
__global__ __launch_bounds__(256) void loss1_kernel(const float* __restrict__ xp,
                                                    const float* __restrict__ xg,
                                                    const float* __restrict__ w,
                                                    float* ws) {
    int b = blockIdx.x >> 3;
    int n = ((blockIdx.x & 7) << 8) + threadIdx.x;
    const float* base = ws + WS_BATCH0 + b * WS_BSTRIDE;
    float R[9], mug[3], mup[3];
    for (int k = 0; k < 9; k++) R[k] = base[O_R + k];
    for (int k = 0; k < 3; k++) { mug[k] = base[O_MUG + k]; mup[k] = base[O_MUP + k]; }

    const float* p = xp + ((size_t)b * N_DIM + n) * 3;
    const float* g = xg + ((size_t)b * N_DIM + n) * 3;
    float wv = w[(size_t)b * N_DIM + n];
    float gc[3] = {g[0] - mug[0], g[1] - mug[1], g[2] - mug[2]};
    float diff[3];
    for (int i = 0; i < 3; i++) {
        float xa = R[3*i+0]*gc[0] + R[3*i+1]*gc[1] + R[3*i+2]*gc[2] + mup[i];
        diff[i] = p[i] - xa;
    }
    float sq = diff[0]*diff[0] + diff[1]*diff[1] + diff[2]*diff[2];
    float val = wv * ((sq > 0.0f) ? __builtin_amdgcn_sqrtf(sq) : 0.0f);

    __shared__ float sd[256];
    sd[threadIdx.x] = val;
    __syncthreads();
    for (int off = 128; off > 0; off >>= 1) {
        if (threadIdx.x < off) sd[threadIdx.x] += sd[threadIdx.x + off];
        __syncthreads();
    }
    if (threadIdx.x == 0) atomicAdd(&ws[WS_LOSS1], sd[0]);
}

// Bond term: one wave = one 16x16 pair tile. Gram matrices via V_WMMA_F32_16X16X4_F32.
// d_ij^2 = |xi|^2 + |xj|^2 - 2*G_ij ; accumulate w_i*w_j*(d_pred - d_gt)^2.
__global__ __launch_bounds__(256) void bond_kernel(const float* __restrict__ xp,
                                                   const float* __restrict__ xg,
                                                   const float* __restrict__ w,
                                                   float* ws) {
    int tid  = blockIdx.x * 256 + threadIdx.x;
    int wave = tid >> 5;
    int lane = tid & 31;
    int b   = wave >> 14;            // / TILES_PER_B (16384)
    int rem = wave & 16383;
    int ti  = rem >> 7;              // / NT (128)
    int tj  = rem & 127;
    int r   = lane & 15;
    int hi  = lane >> 4;

    const float* xpb = xp + (size_t)b * N_DIM * 3;
    const float* xgb = xg + (size_t)b * N_DIM * 3;
    const float* wb  = w  + (size_t)b * N_DIM;
    int irow = ti * 16 + r;
    int jrow = tj * 16 + r;

    float pix = xpb[irow*3+0], piy = xpb[irow*3+1], piz = xpb[irow*3+2];
    float gix = xgb[irow*3+0], giy = xgb[irow*3+1], giz = xgb[irow*3+2];
    float pjx = xpb[jrow*3+0], pjy = xpb[jrow*3+1], pjz = xpb[jrow*3+2];
    float gjx = xgb[jrow*3+0], gjy = xgb[jrow*3+1], gjz = xgb[jrow*3+2];
    float wi = wb[irow], wj = wb[jrow];

    float nip = pix*pix + piy*piy + piz*piz;
    float nig = gix*gix + giy*giy + giz*giz;
    float njp = pjx*pjx + pjy*pjy + pjz*pjz;
    float njg = gjx*gjx + gjy*gjy + gjz*gjz;

    // A 16x4 layout: lanes 0-15 hold (K=0,K=1) = (x,y); lanes 16-31 hold (K=2,K=3) = (z,0)
    v2f ap, bp, ag, bg;
    ap.x = hi ? piz : pix;  ap.y = hi ? 0.0f : piy;
    bp.x = hi ? pjz : pjx;  bp.y = hi ? 0.0f : pjy;
    ag.x = hi ? giz : gix;  ag.y = hi ? 0.0f : giy;
    bg.x = hi ? gjz : gjx;  bg.y = hi ? 0.0f : gjy;

    v8f cz = {0.f, 0.f, 0.f, 0.f, 0.f, 0.f, 0.f, 0.f};
    v8f Gp = __builtin_amdgcn_wmma_f32_16x16x4_f32(false, ap, false, bp,
                                                   (short)0, cz, false, false);
    v8f Gg = __builtin_amdgcn_wmma_f32_16x16x4_f32(false, ag, false, bg,
                                                   (short)0, cz, false, false);

    // C/D layout: VGPR v, lane L -> (M = v + 8*(L>=16), N = L%16)
    float acc = 0.0f;
#pragma unroll
    for (int v = 0; v < 8; ++v) {
        int m = v + (hi << 3);
        float m_nip = __shfl(nip, m, 32);
        float m_nig = __shfl(nig, m, 32);
        float m_wi  = __shfl(wi,  m, 32);
        float dp2 = fmaxf(m_nip + njp - 2.0f * Gp[v], 0.0f);
        float dg2 = fmaxf(m_nig + njg - 2.0f * Gg[v], 0.0f);
        float d = __builtin_amdgcn_sqrtf(dp2) - __builtin_amdgcn_sqrtf(dg2);
        acc = fmaf(m_wi * wj, d * d, acc);
    }
    for (int off = 16; off > 0; off >>= 1) acc += __shfl_xor(acc, off, 32);
    if (lane == 0) atomicAdd(&ws[WS_BOND], acc);
}

__global__ void finalize_kernel(const float* __restrict__ ht, const float* __restrict__ ws,
                                float* __restrict__ out) {
    int b = threadIdx.x;
    if (b >= B_DIM) return;
    float tw = 0.0f, tw2 = 0.0f;
    for (int bb = 0; bb < B_DIM; ++bb) {
        float s = ws[WS_BATCH0 + bb * WS_BSTRIDE + O_SW];
        tw += s;
        tw2 += s * s;   // sum over pairs of w_i*w_j = (sum w)^2 per batch
    }
    float loss = ws[WS_LOSS1] / tw + ALPHA_BOND * ws[WS_BOND] / tw2;
    float h = ht[b];
    float coef = (h * h + SIGMA_DATA * SIGMA_DATA) /
                 ((h + SIGMA_DATA) * (h + SIGMA_DATA));
    out[b] = coef * loss;
}

extern "C" void kernel_launch(void* const* d_in, const int* in_sizes, int n_in,
                              void* d_out, int out_size, void* d_ws, size_t ws_size,
                              hipStream_t stream) {
    const float* xp = (const float*)d_in[0];   // [B,N,3]
    const float* xg = (const float*)d_in[1];   // [B,N,3]
    const float* ht = (const float*)d_in[2];   // [B]
    const float* w  = (const float*)d_in[3];   // [B,N]
    float* out = (float*)d_out;
    float* ws  = (float*)d_ws;

    init_ws<<<1, 256, 0, stream>>>(ws);
    stats_kernel<<<B_DIM * (N_DIM / 256), 256, 0, stream>>>(xp, xg, w, ws);
    svd_kernel<<<1, 32, 0, stream>>>(ws);
    loss1_kernel<<<B_DIM * (N_DIM / 256), 256, 0, stream>>>(xp, xg, w, ws);
    int bond_blocks = (B_DIM * TILES_PER_B * 32) / 256;   // 8192
    bond_kernel<<<bond_blocks, 256, 0, stream>>>(xp, xg, w, ws);
    finalize_kernel<<<1, 32, 0, stream>>>(ht, ws, out);
}